// BaseLayer_67568425501388
// MI455X (gfx1250) — compile-verified
//
#include <hip/hip_runtime.h>
#include <stdint.h>

// ---------------------------------------------------------------------------
// Sorted ragged segment mean+sum. HBM-bound: 1.07GB stream / 23.3 TB/s ~ 47us.
// CDNA5 paths used:
//   * global_load_async_to_lds_b128 + s_wait_asynccnt double buffering
//     (streams 4KB row tiles into LDS without a VGPR round trip)
//   * v_wmma_f32_16x16x4_f32 one-hot segment GEMM: D = A(one-hot 16x4) x
//     B(4 rows x 16 feats) + C. Row m of D = partial sum of local segment m;
//     C row 0 carries the open segment across tiles, so atomics fire ~once
//     per segment (all L2-resident: 17MB << 192MB L2).
// ---------------------------------------------------------------------------

typedef __attribute__((ext_vector_type(2))) float v2f;
typedef __attribute__((ext_vector_type(8))) float v8f;

#define D_FEAT   64                  // features per row
#define TILE_R   16                  // rows per LDS tile (16*256B = 4KB)
#define NTILES   16                  // tiles per wave-chunk
#define CHUNK    (TILE_R * NTILES)   // 256 rows per wave
#define WPB      8                   // waves per block (256 threads, wave32)

// Issue one 16-row (4KB) tile: 8 async b128 ops, 32 lanes x 16B = 512B each.
// INST_OFFSET is added to BOTH the LDS and global addresses (ISA 08), so one
// base pair covers the contiguous tile with compile-time offsets.
static __device__ __forceinline__ void issue_tile_async(const float* gp,
                                                        uint32_t ldsa) {
#define ALD(OFF)                                                               \
  asm volatile("global_load_async_to_lds_b128 %0, %1, off offset:" OFF         \
               :: "v"(ldsa), "v"(gp) : "memory")
  ALD("0");    ALD("512");  ALD("1024"); ALD("1536");
  ALD("2048"); ALD("2560"); ALD("3072"); ALD("3584");
#undef ALD
}

// Select element i (0..7) of a WMMA accumulator (uniform i; rare path).
static __device__ __forceinline__ float vsel8(v8f c, int i) {
  float r = c[0];
  r = (i == 1) ? c[1] : r;
  r = (i == 2) ? c[2] : r;
  r = (i == 3) ? c[3] : r;
  r = (i == 4) ? c[4] : r;
  r = (i == 5) ? c[5] : r;
  r = (i == 6) ? c[6] : r;
  r = (i == 7) ? c[7] : r;
  return r;
}

// Carry layout: lanes 0-15 hold feature fg*16+lane for fg=0..3.
static __device__ __forceinline__ void flush_carry(float* __restrict__ out,
                                                   float* __restrict__ counts,
                                                   int sid, const float cr[4],
                                                   int cnt, int lane) {
  if (lane < 16) {
    float* dst = out + (size_t)sid * 128 + 64 + lane;   // sum half of [G,128]
    atomicAdd(dst + 0,  cr[0]);
    atomicAdd(dst + 16, cr[1]);
    atomicAdd(dst + 32, cr[2]);
    atomicAdd(dst + 48, cr[3]);
  }
  if (lane == 0) atomicAdd(&counts[sid], (float)cnt);
}

__global__ void __launch_bounds__(32 * WPB)
seg_reduce_kernel(const float* __restrict__ nbr, const int* __restrict__ seg,
                  float* __restrict__ out, float* __restrict__ counts,
                  long long nrows) {
  __shared__ __align__(16) float lds[WPB][2][TILE_R * D_FEAT];  // 64KB/block

  const int lane = threadIdx.x & 31;
  const int n15  = lane & 15;
  const int wave = threadIdx.x >> 5;
  const long long chunk_start =
      ((long long)blockIdx.x * WPB + wave) * (long long)CHUNK;
  if (chunk_start >= nrows) return;   // wave-uniform; no block barriers used

  long long remll = nrows - chunk_start;
  const int rows  = remll < (long long)CHUNK ? (int)remll : CHUNK;
  const int nfull = rows / TILE_R;

  const float* gbase = nbr + chunk_start * D_FEAT + lane * 4;
  // Low 32 bits of a flat LDS pointer are the local byte offset.
  const uint32_t lds_a =
      (uint32_t)(uintptr_t)(&lds[wave][0][0]) + (uint32_t)(lane * 16);
  const uint32_t lds_b = lds_a + (uint32_t)(TILE_R * D_FEAT * 4);

  if (nfull > 0) issue_tile_async(gbase, lds_a);   // prefetch tile 0

  float carry[4] = {0.f, 0.f, 0.f, 0.f};  // open-segment partial (lanes 0-15)
  int carry_cnt = 0;
  int cur_id = seg[chunk_start];          // wave-uniform

  for (int t = 0; t < nfull; ++t) {
    if (t + 1 < nfull) {
      issue_tile_async(gbase + (size_t)(t + 1) * TILE_R * D_FEAT,
                       ((t + 1) & 1) ? lds_b : lds_a);
      asm volatile("s_wait_asynccnt 8" ::: "memory");  // tile t complete
    } else {
      asm volatile("s_wait_asynccnt 0" ::: "memory");
    }
    const float* buf = &lds[wave][t & 1][0];

    // --- per-row segment ids (upper half-wave mirrors lower) ---
    int myid = seg[chunk_start + (long long)t * TILE_R + n15];
    int pid  = __shfl(myid, (n15 == 0) ? 0 : (n15 - 1), 32);
    bool bd  = (myid != ((n15 == 0) ? cur_id : pid));
    unsigned bmask = (unsigned)__ballot(bd) & 0xFFFFu;
    int m     = __popc(bmask & ((2u << n15) - 1));  // local segment of row n15
    int mlast = __popc(bmask);                      // local segment of row 15

    // Row 0 starts a new segment -> carried segment is complete: flush now so
    // local indices fit 0..15 (<=15 boundaries remain among rows 1..15).
    if (bmask & 1u) {
      flush_carry(out, counts, cur_id, carry, carry_cnt, lane);
      carry[0] = carry[1] = carry[2] = carry[3] = 0.f;
      carry_cnt = 0;
      m -= 1;
      mlast -= 1;
    }

    // --- accumulators: row 0 seeded with the carried segment ---
    v8f c0 = {}, c1 = {}, c2 = {}, c3 = {};
    c0[0] = (lane < 16) ? carry[0] : 0.f;
    c1[0] = (lane < 16) ? carry[1] : 0.f;
    c2[0] = (lane < 16) ? carry[2] : 0.f;
    c3[0] = (lane < 16) ? carry[3] : 0.f;

    // --- one-hot segment GEMM: 4 K-steps x 4 feature groups ---
#pragma unroll
    for (int kb = 0; kb < 4; ++kb) {
      // A 16x4 f32 layout: lanes 0-15 hold K=0,1 ; lanes 16-31 hold K=2,3.
      int ra = 4 * kb + ((lane & 16) >> 3);
      int rb = ra + 1;
      int ma = __shfl(m, ra, 32);
      int mb = __shfl(m, rb, 32);
      v2f a;
      a[0] = (ma == n15) ? 1.f : 0.f;
      a[1] = (mb == n15) ? 1.f : 0.f;
      // B 4x16 f32 layout: VGPR0 = rows {0,2}, VGPR1 = rows {1,3}, n = lane%16.
      const float* pa = buf + ra * D_FEAT + n15;
      const float* pb = buf + rb * D_FEAT + n15;
      v2f b0 = {pa[0],  pb[0]};
      v2f b1 = {pa[16], pb[16]};
      v2f b2 = {pa[32], pb[32]};
      v2f b3 = {pa[48], pb[48]};
      c0 = __builtin_amdgcn_wmma_f32_16x16x4_f32(false, a, false, b0,
                                                 (short)0, c0, false, false);
      c1 = __builtin_amdgcn_wmma_f32_16x16x4_f32(false, a, false, b1,
                                                 (short)0, c1, false, false);
      c2 = __builtin_amdgcn_wmma_f32_16x16x4_f32(false, a, false, b2,
                                                 (short)0, c2, false, false);
      c3 = __builtin_amdgcn_wmma_f32_16x16x4_f32(false, a, false, b3,
                                                 (short)0, c3, false, false);
    }

    // --- flush finished local segments 0..mlast-1 (uniform loop, ~0-1 iters)
    for (int mm = 0; mm < mlast; ++mm) {
      unsigned sm = (unsigned)__ballot(m == mm) & 0xFFFFu;
      int rfirst = __ffs(sm) - 1;
      int sid = __shfl(myid, rfirst, 32);
      int cnt = __popc(sm) + ((mm == 0) ? carry_cnt : 0);
      float f0 = vsel8(c0, mm & 7);
      float f1 = vsel8(c1, mm & 7);
      float f2 = vsel8(c2, mm & 7);
      float f3 = vsel8(c3, mm & 7);
      // D 16x16 layout: rows 0-7 on lanes 0-15 (VGPR m), rows 8-15 on 16-31.
      bool act = (mm < 8) ? (lane < 16) : (lane >= 16);
      if (act) {
        float* dst = out + (size_t)sid * 128 + 64 + n15;
        atomicAdd(dst + 0,  f0);
        atomicAdd(dst + 16, f1);
        atomicAdd(dst + 32, f2);
        atomicAdd(dst + 48, f3);
      }
      if (lane == 0) atomicAdd(&counts[sid], (float)cnt);
    }

    // --- new carry = local segment mlast ---
    {
      float f0 = vsel8(c0, mlast & 7);
      float f1 = vsel8(c1, mlast & 7);
      float f2 = vsel8(c2, mlast & 7);
      float f3 = vsel8(c3, mlast & 7);
      if (mlast >= 8) {   // uniform: move lanes 16-31 -> 0-15
        f0 = __shfl(f0, lane + 16, 32);
        f1 = __shfl(f1, lane + 16, 32);
        f2 = __shfl(f2, lane + 16, 32);
        f3 = __shfl(f3, lane + 16, 32);
      }
      carry[0] = f0; carry[1] = f1; carry[2] = f2; carry[3] = f3;
      int cml = __popc((unsigned)__ballot(m == mlast) & 0xFFFFu);
      carry_cnt = cml + ((mlast == 0) ? carry_cnt : 0);
      cur_id = __shfl(myid, 15, 32);
    }
  }

  // Remainder rows (nrows % TILE_R, not hit for 4M rows): direct loads.
  for (int r = nfull * TILE_R; r < rows; ++r) {
    long long row = chunk_start + r;
    int sid = seg[row];
    if (sid != cur_id) {
      flush_carry(out, counts, cur_id, carry, carry_cnt, lane);
      carry[0] = carry[1] = carry[2] = carry[3] = 0.f;
      carry_cnt = 0;
      cur_id = sid;
    }
    if (lane < 16) {
      const float* src = nbr + row * D_FEAT + n15;
      carry[0] += src[0];
      carry[1] += src[16];
      carry[2] += src[32];
      carry[3] += src[48];
    }
    carry_cnt++;
  }

  flush_carry(out, counts, cur_id, carry, carry_cnt, lane);
}

__global__ void __launch_bounds__(256)
zero_kernel(float* __restrict__ out, float* __restrict__ counts, int G) {
  int i = blockIdx.x * blockDim.x + threadIdx.x;
  int n_out = G * 128;
  if (i < n_out) {
    out[i] = 0.f;
  } else if (i < n_out + G) {
    counts[i - n_out] = 0.f;
  }
}

__global__ void __launch_bounds__(256)
finalize_kernel(float* __restrict__ out, const float* __restrict__ counts,
                int G) {
  int i = blockIdx.x * blockDim.x + threadIdx.x;
  if (i >= G * D_FEAT) return;
  int g = i >> 6;
  int f = i & 63;
  float s = out[(size_t)g * 128 + 64 + f];
  float c = counts[g];
  c = c > 1.f ? c : 1.f;
  out[(size_t)g * 128 + f] = s / c;
}

extern "C" void kernel_launch(void* const* d_in, const int* in_sizes, int n_in,
                              void* d_out, int out_size, void* d_ws,
                              size_t ws_size, hipStream_t stream) {
  const float* nbr = (const float*)d_in[0];
  const int* seg = (const int*)d_in[1];
  float* out = (float*)d_out;
  float* counts = (float*)d_ws;   // G floats of scratch

  const long long nrows = (long long)in_sizes[0] / D_FEAT;
  const int G = out_size / 128;

  {
    int n = G * 128 + G;
    zero_kernel<<<(n + 255) / 256, 256, 0, stream>>>(out, counts, G);
  }
  {
    long long nwaves = (nrows + CHUNK - 1) / CHUNK;
    int nblocks = (int)((nwaves + WPB - 1) / WPB);
    seg_reduce_kernel<<<nblocks, 32 * WPB, 0, stream>>>(nbr, seg, out, counts,
                                                        nrows);
  }
  {
    int n = G * D_FEAT;
    finalize_kernel<<<(n + 255) / 256, 256, 0, stream>>>(out, counts, G);
  }
}